// LossFunctions_86397562126683
// MI455X (gfx1250) — compile-verified
//
#include <hip/hip_runtime.h>
#include <math.h>

typedef float v2f __attribute__((ext_vector_type(2)));
typedef float v8f __attribute__((ext_vector_type(8)));

#define WAVES_PER_BLOCK 4
#define ROWS_PER_WAVE 2
#define NSLOT 7
#define DDIM 128

struct __align__(16) WaveScratch {
  float ai_c[2][8][32];   // [row][slot(7=zero pad)][k in chunk]
  float ah_c[2][8][32];
  float o_c [2][8][32];
  float r_c [2][8][32];
  float m_c [2][32];
  float nrm [4][16];      // 0:ai 1:ah 2:o 3:r ; idx row*8+slot
  float cost[4][64];      // 0:row0 ai-ah, 1:row1 ai-ah, 2:row0 o-r, 3:row1 o-r ; i*8+j
  float red_c[32];
  int   red_i[32];
  float mt[2];
  float wacc;
  float pad_;
};

// Decode lexicographic permutation index p in [0,5040) (matches itertools order)
// and sum cost[i][perm[i]].
__device__ __forceinline__ float permCost(const float* C, int p) {
  unsigned avail = 0x6543210u;
  int rem = p;
  float s = 0.f;
#pragma unroll
  for (int i = 0; i < 6; ++i) {
    int f = (i==0)?720:(i==1)?120:(i==2)?24:(i==3)?6:(i==4)?2:1;
    int q = rem / f; rem -= q * f;
    int sh = q * 4;
    int sel = (avail >> sh) & 0xF;
    avail = (avail & ((1u << sh) - 1u)) | ((avail >> (sh + 4)) << sh);
    s += C[i * 8 + sel];
  }
  s += C[48 + (avail & 0xF)];
  return s;
}

__global__ void __launch_bounds__(128, 1)
loss_kernel(const float* __restrict__ ai, const float* __restrict__ ah,
            const float* __restrict__ mu, const float* __restrict__ lv,
            const float* __restrict__ o,  const float* __restrict__ mask,
            const float* __restrict__ gamma, float* __restrict__ out)
{
  __shared__ WaveScratch ws[WAVES_PER_BLOCK];
  const int lane = threadIdx.x & 31;
  const int wave = threadIdx.x >> 5;
  WaveScratch& W = ws[wave];
  const int b0 = (blockIdx.x * WAVES_PER_BLOCK + wave) * ROWS_PER_WAVE;

  // ---- init: zero pad slot 7 of each staging array, mt, wacc ----
  for (int t = lane; t < 64; t += 32) {
    int r = t >> 5, k = t & 31;
    W.ai_c[r][7][k] = 0.f; W.ah_c[r][7][k] = 0.f;
    W.o_c [r][7][k] = 0.f; W.r_c [r][7][k] = 0.f;
  }
  if (lane < 2) W.mt[lane] = 0.f;
  if (lane == 0) W.wacc = 0.f;
  __syncthreads();

  v8f acc_g  = {0.f,0.f,0.f,0.f,0.f,0.f,0.f,0.f};  // ai . ah^T
  v8f acc_q  = acc_g;                               // o  . r^T
  v8f acc_na = acc_g, acc_nb = acc_g, acc_no = acc_g, acc_nr = acc_g; // self-Grams (diag = norms)
  float part = 0.f;  // per-lane: reorder + entropy + (-2)*KL partials

  const int m16  = lane & 15;
  const int kh   = (lane >> 4) << 1;   // f32 WMMA frag: lanes16-31 hold K+2,K+3
  const int frow = m16 >> 3, fslot = m16 & 7;

  for (int c = 0; c < 4; ++c) {
    const int k0 = c * 32;
    // ---- stage ai/ah/o chunk (coalesced float4) ----
    for (int t = lane; t < 112; t += 32) {
      int r = t / 56, rem = t % 56, s = rem >> 3, q = rem & 7;
      long base = (long)((b0 + r) * NSLOT + s) * DDIM + k0 + q * 4;
      float4 va = *(const float4*)(ai + base);
      float4 vh = *(const float4*)(ah + base);
      float4 vo = *(const float4*)(o  + base);
      *(float4*)&W.ai_c[r][s][q*4] = va;
      *(float4*)&W.ah_c[r][s][q*4] = vh;
      *(float4*)&W.o_c [r][s][q*4] = vo;
    }
    // ---- stage mask chunk + entropy + mask_term ----
    if (lane < 16) {
      int r = lane >> 3, q = lane & 7;
      float4 vm = *(const float4*)(mask + (long)(b0 + r) * DDIM + k0 + q * 4);
      *(float4*)&W.m_c[r][q*4] = vm;
      float e = 0.f, mtp = 0.f;
      e -= vm.x * __logf(vm.x + 1e-10f); mtp += 1.f - vm.x;
      e -= vm.y * __logf(vm.y + 1e-10f); mtp += 1.f - vm.y;
      e -= vm.z * __logf(vm.z + 1e-10f); mtp += 1.f - vm.z;
      e -= vm.w * __logf(vm.w + 1e-10f); mtp += 1.f - vm.w;
      part += e;
      atomicAdd(&W.mt[r], mtp);
    }
    __syncthreads();
    // ---- r = ah / mask ----
    for (int t = lane; t < 112; t += 32) {
      int r = t / 56, rem = t % 56, s = rem >> 3, q = rem & 7;
      float4 vh = *(float4*)&W.ah_c[r][s][q*4];
      float4 vm = *(float4*)&W.m_c[r][q*4];
      float4 vr;
      vr.x = vh.x / vm.x; vr.y = vh.y / vm.y;
      vr.z = vh.z / vm.z; vr.w = vh.w / vm.w;
      *(float4*)&W.r_c[r][s][q*4] = vr;
    }
    // ---- reorder loss partial: sum ||ah[s]-ah[s-1]||^2 ----
    for (int t = lane; t < 96; t += 32) {
      int r = t / 48, rem = t % 48, s = (rem >> 3) + 1, q = rem & 7;
      float4 a = *(float4*)&W.ah_c[r][s][q*4];
      float4 b = *(float4*)&W.ah_c[r][s-1][q*4];
      float dx = a.x-b.x, dy = a.y-b.y, dz = a.z-b.z, dw = a.w-b.w;
      part += dx*dx + dy*dy + dz*dz + dw*dw;
    }
    __syncthreads();
    // ---- WMMA: 6 Grams, K=32 chunk in 8 steps of K=4 ----
    // f32 A-frag 16x4: lane(m=l&15): VGPR0=A[m][k+kh], VGPR1=A[m][k+kh+1].
    // B-frag mirrors with n=l&15, so the same v2f serves as A and B of X.X^T.
#pragma unroll
    for (int st = 0; st < 8; ++st) {
      int ko = st * 4 + kh;
      v2f fa = *(const v2f*)&W.ai_c[frow][fslot][ko];
      v2f fb = *(const v2f*)&W.ah_c[frow][fslot][ko];
      v2f fo = *(const v2f*)&W.o_c [frow][fslot][ko];
      v2f fr = *(const v2f*)&W.r_c [frow][fslot][ko];
      acc_g  = __builtin_amdgcn_wmma_f32_16x16x4_f32(false, fa, false, fb, (short)0, acc_g,  false, false);
      acc_na = __builtin_amdgcn_wmma_f32_16x16x4_f32(false, fa, false, fa, (short)0, acc_na, false, false);
      acc_nb = __builtin_amdgcn_wmma_f32_16x16x4_f32(false, fb, false, fb, (short)0, acc_nb, false, false);
      acc_q  = __builtin_amdgcn_wmma_f32_16x16x4_f32(false, fo, false, fr, (short)0, acc_q,  false, false);
      acc_no = __builtin_amdgcn_wmma_f32_16x16x4_f32(false, fo, false, fo, (short)0, acc_no, false, false);
      acc_nr = __builtin_amdgcn_wmma_f32_16x16x4_f32(false, fr, false, fr, (short)0, acc_nr, false, false);
    }
    __syncthreads();  // protect LDS before next chunk overwrites it
  }

  // ---- KL streaming: -2 * sum(1 + lv - mu^2 - exp(lv)) ----
  {
    float kl = 0.f;
    const long base = (long)b0 * NSLOT * 32;  // float4 units
    const float4* mu4 = (const float4*)mu;
    const float4* lv4 = (const float4*)lv;
    for (int t = lane; t < 2 * NSLOT * 32; t += 32) {
      float4 m = mu4[base + t];
      float4 l = lv4[base + t];
      kl += 1.f + l.x - (m.x*m.x + __expf(l.x));
      kl += 1.f + l.y - (m.y*m.y + __expf(l.y));
      kl += 1.f + l.z - (m.z*m.z + __expf(l.z));
      kl += 1.f + l.w - (m.w*m.w + __expf(l.w));
    }
    part += -2.0f * kl;
  }
  atomicAdd(&W.wacc, part);

  // ---- norms from self-Gram diagonals: C[d][d] -> vgpr d&7, lane (d&7)+(d>=8?16+8?..)
  // d<8: vgpr d, lane d ; d>=8: vgpr d-8, lane (d-8)+8+16 = d+16
#pragma unroll
  for (int v = 0; v < 7; ++v) {
    if (lane == v) {
      W.nrm[0][v] = acc_na[v]; W.nrm[1][v] = acc_nb[v];
      W.nrm[2][v] = acc_no[v]; W.nrm[3][v] = acc_nr[v];
    }
    if (lane == 24 + v) {
      W.nrm[0][8+v] = acc_na[v]; W.nrm[1][8+v] = acc_nb[v];
      W.nrm[2][8+v] = acc_no[v]; W.nrm[3][8+v] = acc_nr[v];
    }
  }
  __syncthreads();

  // ---- cost matrices: sqrt(max(||x||^2 + ||y||^2 - 2 x.y, 0)) ----
#pragma unroll
  for (int v = 0; v < 7; ++v) {
    float g = acc_g[v], qq = acc_q[v];
    if (lane < 7) {  // row0 block: C[v][lane]
      W.cost[0][v*8+lane] = sqrtf(fmaxf(W.nrm[0][v] + W.nrm[1][lane] - 2.f*g,  0.f));
      W.cost[2][v*8+lane] = sqrtf(fmaxf(W.nrm[2][v] + W.nrm[3][lane] - 2.f*qq, 0.f));
    }
    if (lane >= 24 && lane < 31) {  // row1 block: C[8+v][8+j], j = lane-24
      int j = lane - 24;
      W.cost[1][v*8+j] = sqrtf(fmaxf(W.nrm[0][8+v] + W.nrm[1][8+j] - 2.f*g,  0.f));
      W.cost[3][v*8+j] = sqrtf(fmaxf(W.nrm[2][8+v] + W.nrm[3][8+j] - 2.f*qq, 0.f));
    }
  }
  __syncthreads();

  // ---- brute-force optimal assignment (5040 perms, 158/lane) x 4 matrices ----
  float rec = 0.f, cond0 = 0.f, cond1 = 0.f;
  for (int cm = 0; cm < 4; ++cm) {
    const float* C = W.cost[cm];
    float bc = 3.4e38f; int bp = 0;
    for (int p = lane; p < 5040; p += 32) {
      float s = permCost(C, p);
      if (s < bc) { bc = s; bp = p; }  // increasing p => first-min tie-break
    }
    W.red_c[lane] = bc; W.red_i[lane] = bp;
    __syncthreads();
    if (lane == 0) {
      bc = W.red_c[0]; bp = W.red_i[0];
      for (int l = 1; l < 32; ++l) {
        float c2 = W.red_c[l]; int p2 = W.red_i[l];
        if (c2 < bc || (c2 == bc && p2 < bp)) { bc = c2; bp = p2; }
      }
      // re-decode winner, accumulate matched losses (0.5*cost^2 = 0.5*||diff||^2)
      unsigned avail = 0x6543210u; int rem = bp;
#pragma unroll
      for (int i = 0; i < 7; ++i) {
        int f = (i==0)?720:(i==1)?120:(i==2)?24:(i==3)?6:(i==4)?2:1;
        int q = rem / f; rem -= q * f;
        int sh = q * 4;
        int sel = (avail >> sh) & 0xF;
        avail = (avail & ((1u << sh) - 1u)) | ((avail >> (sh + 4)) << sh);
        float cst = C[i*8 + sel];
        if (cm < 2) rec += 0.5f * cst * cst;
        else {
          float t = fabsf(0.5f * cst * cst - gamma[sel]);
          if (cm == 2) cond0 += t; else cond1 += t;
        }
      }
    }
    __syncthreads();
  }

  if (lane == 0) {
    float total = W.wacc + rec + W.mt[0] * cond0 + W.mt[1] * cond1;
    atomicAdd(out, total);
  }
}

__global__ void zero_out_kernel(float* out) {
  if (threadIdx.x == 0 && blockIdx.x == 0) out[0] = 0.f;
}

extern "C" void kernel_launch(void* const* d_in, const int* in_sizes, int n_in,
                              void* d_out, int out_size, void* d_ws, size_t ws_size,
                              hipStream_t stream) {
  const float* ai    = (const float*)d_in[0];
  const float* ah    = (const float*)d_in[1];
  const float* mu    = (const float*)d_in[2];
  const float* lv    = (const float*)d_in[3];
  const float* o     = (const float*)d_in[4];
  const float* mask  = (const float*)d_in[5];
  const float* gamma = (const float*)d_in[6];
  float* out = (float*)d_out;

  zero_out_kernel<<<1, 32, 0, stream>>>(out);
  // 8192 rows / (4 waves * 2 rows) = 1024 blocks of 128 threads
  loss_kernel<<<1024, 128, 0, stream>>>(ai, ah, mu, lv, o, mask, gamma, out);
}